// EE_Conv_88880053223551
// MI455X (gfx1250) — compile-verified
//
#include <hip/hip_runtime.h>
#include <math.h>

typedef __attribute__((ext_vector_type(2))) float v2f;
typedef __attribute__((ext_vector_type(8))) float v8f;

#define FEAT 64

// ---------------------------------------------------------------------------
// 1) Node transform  T = X @ W^T  via V_WMMA_F32_16X16X4_F32 (wave32).
//    One wave owns a 16-row tile of X; A fragments (16 k-steps of K=4) are
//    preloaded once and reused across the 4 sixteen-column output tiles.
// ---------------------------------------------------------------------------
__global__ void node_gemm_wmma(const float* __restrict__ X,   // [N, 64]
                               const float* __restrict__ W,   // [64, 64] row-major [O][I]
                               float* __restrict__ T,         // [N, 64]
                               int N)
{
  const int lane  = threadIdx.x & 31;
  const int wave  = blockIdx.x * (blockDim.x >> 5) + (threadIdx.x >> 5);
  const int m0    = wave << 4;
  if (m0 >= N) return;                       // wave-uniform: EXEC stays all-ones

  const int r     = lane & 15;               // A: row in tile / B: col in tile
  const int khalf = (lane >> 4) << 1;        // lanes 0-15 -> K{0,1}, 16-31 -> K{2,3}

  // Preload all 16 A fragments (16x64 tile of X), 2 floats per lane each.
  v2f a[16];
  const float* xrow = X + (size_t)(m0 + r) * FEAT;
#pragma unroll
  for (int ks = 0; ks < 16; ++ks) {
    const int kk = (ks << 2) + khalf;
    a[ks].x = xrow[kk];
    a[ks].y = xrow[kk + 1];
  }

#pragma unroll
  for (int ct = 0; ct < 4; ++ct) {
    const int o0 = ct << 4;
    const float* wrow = W + (size_t)(o0 + r) * FEAT;   // B[k][o] = W[o][k]
    v8f acc = {};
#pragma unroll
    for (int ks = 0; ks < 16; ++ks) {
      const int kk = (ks << 2) + khalf;
      v2f b;
      b.x = wrow[kk];
      b.y = wrow[kk + 1];
      // (neg_a, A, neg_b, B, c_mod, C, reuse_a, reuse_b)
      acc = __builtin_amdgcn_wmma_f32_16x16x4_f32(false, a[ks], false, b,
                                                  (short)0, acc, false, false);
    }
    // C/D layout: acc[j] = (M = j + 8*(lane>=16), N = lane&15)
    const int mhi = (lane >> 4) << 3;
#pragma unroll
    for (int j = 0; j < 8; ++j) {
      T[(size_t)(m0 + mhi + j) * FEAT + (o0 + r)] = acc[j];
    }
  }
}

// ---------------------------------------------------------------------------
// 2) Init running max to -inf (float4 stores).
// ---------------------------------------------------------------------------
__global__ void fill_neg_inf4(float4* __restrict__ p, long long n4)
{
  const long long i = (long long)blockIdx.x * blockDim.x + threadIdx.x;
  const float ninf = -__builtin_inff();
  if (i < n4) p[i] = make_float4(ninf, ninf, ninf, ninf);
}

// ---------------------------------------------------------------------------
// 3) Edge scatter-max:  g[dst] = max(g[dst], d_e * T[src])   (per feature)
//    One wave per edge: indices scalarized via readfirstlane -> s_load;
//    each lane covers 2 features (b64 load + 2 native f32 max atomics).
//    CDNA5 GLOBAL_ATOMIC_MAX_NUM_F32: non-returning, IEEE maximumNumber,
//    order-independent => bitwise deterministic result.
// ---------------------------------------------------------------------------
__device__ __forceinline__ void atomic_max_f32(float* addr, float val)
{
  asm volatile("global_atomic_max_num_f32 %0, %1, off"
               :
               : "v"(addr), "v"(val)
               : "memory");
}

__global__ void edge_scatter_max(const float* __restrict__ T,
                                 const float* __restrict__ edge_d,
                                 const int* __restrict__ src,
                                 const int* __restrict__ dst,
                                 float* __restrict__ g,
                                 int E)
{
  const int wave = blockIdx.x * (blockDim.x >> 5) + (threadIdx.x >> 5);
  if (wave >= E) return;                       // wave-uniform exit
  const int lane = threadIdx.x & 31;

  // Force edge index into an SGPR so the index/weight loads become scalar.
  const int e = __builtin_amdgcn_readfirstlane(wave);
  const float d = edge_d[e];                   // s_load_b32
  const int   s = src[e];                      // s_load_b32
  const int   t = dst[e];                      // s_load_b32

  const float2 tv = *(const float2*)(T + (size_t)s * FEAT + (lane << 1));
  float* gp = g + (size_t)t * FEAT + (lane << 1);
  atomic_max_f32(gp,     d * tv.x);
  atomic_max_f32(gp + 1, d * tv.y);
}

// ---------------------------------------------------------------------------
// 4) Column means of node_feat (64 values) -- feeds the analytic mean of P.
// ---------------------------------------------------------------------------
__global__ void colmean_X(const float* __restrict__ X, float* __restrict__ cm, int N)
{
  __shared__ float sm[256];
  const int c = blockIdx.x;
  float s = 0.0f;
  for (int n = threadIdx.x; n < N; n += blockDim.x)
    s += X[(size_t)n * FEAT + c];
  sm[threadIdx.x] = s;
  __syncthreads();
  for (int off = 128; off > 0; off >>= 1) {
    if (threadIdx.x < off) sm[threadIdx.x] += sm[threadIdx.x + off];
    __syncthreads();
  }
  if (threadIdx.x == 0) cm[c] = sm[0] / (float)N;
}

// ---------------------------------------------------------------------------
// 5) out[f] = mean_n g[n][f]  +  cm . phi_w[f]  +  theta_b[f] + phi_b[f]
// ---------------------------------------------------------------------------
__global__ void finalize(const float* __restrict__ g,
                         const float* __restrict__ cm,
                         const float* __restrict__ phi_w,
                         const float* __restrict__ theta_b,
                         const float* __restrict__ phi_b,
                         float* __restrict__ out, int N)
{
  __shared__ float sm[256];
  const int f = blockIdx.x;
  float s = 0.0f;
  for (int n = threadIdx.x; n < N; n += blockDim.x)
    s += g[(size_t)n * FEAT + f];
  sm[threadIdx.x] = s;
  __syncthreads();
  for (int off = 128; off > 0; off >>= 1) {
    if (threadIdx.x < off) sm[threadIdx.x] += sm[threadIdx.x + off];
    __syncthreads();
  }
  if (threadIdx.x == 0) {
    float dot = 0.0f;
#pragma unroll
    for (int i = 0; i < FEAT; ++i) dot += cm[i] * phi_w[(size_t)f * FEAT + i];
    out[f] = sm[0] / (float)N + dot + theta_b[f] + phi_b[f];
  }
}

// ---------------------------------------------------------------------------
extern "C" void kernel_launch(void* const* d_in, const int* in_sizes, int n_in,
                              void* d_out, int out_size, void* d_ws, size_t ws_size,
                              hipStream_t stream)
{
  const float* node_feat = (const float*)d_in[0];
  const float* edge_d    = (const float*)d_in[1];
  const float* theta_w   = (const float*)d_in[2];
  const float* theta_b   = (const float*)d_in[3];
  const float* phi_w     = (const float*)d_in[4];
  const float* phi_b     = (const float*)d_in[5];
  const int*   src       = (const int*)d_in[6];
  const int*   dst       = (const int*)d_in[7];

  const int N = in_sizes[0] / FEAT;   // 50000
  const int E = in_sizes[1];          // 800000

  // Workspace: T [N*64] | g [N*64] | cm [64]   (~25.6 MB)
  float* T  = (float*)d_ws;
  float* g  = T + (size_t)N * FEAT;
  float* cm = g + (size_t)N * FEAT;

  // 1) T = X @ theta_w^T  (WMMA f32 16x16x4)
  const int tiles = (N + 15) / 16;                  // 3125, exact
  node_gemm_wmma<<<(tiles + 7) / 8, 256, 0, stream>>>(node_feat, theta_w, T, N);

  // 2) g = -inf
  const long long g4 = ((long long)N * FEAT) / 4;
  fill_neg_inf4<<<(int)((g4 + 255) / 256), 256, 0, stream>>>((float4*)g, g4);

  // 3) scatter-max over edges (one wave per edge, 8 waves per block)
  edge_scatter_max<<<(E + 7) / 8, 256, 0, stream>>>(T, edge_d, src, dst, g, E);

  // 4) column means of X
  colmean_X<<<FEAT, 256, 0, stream>>>(node_feat, cm, N);

  // 5) finalize -> [1, 64]
  finalize<<<FEAT, 256, 0, stream>>>(g, cm, phi_w, theta_b, phi_b,
                                     (float*)d_out, N);
}